// PM25_GNN_3418793967715
// MI455X (gfx1250) — compile-verified
//
#include <hip/hip_runtime.h>
#include <math.h>

#define NB    8
#define NN    1000
#define NP    1008          /* 63*16, row-padded M */
#define HISTN 8
#define PREDN 12
#define INDIM 14
#define HIDG  256
#define HEADS 8
#define GOUT  13
#define HID   64
#define C1    (HEADS*HIDG)   /* 2048 */

typedef __attribute__((ext_vector_type(2))) float v2f;
typedef __attribute__((ext_vector_type(8))) float v8f;

__device__ __forceinline__ v8f wmma4(v2f a, v2f b, v8f c) {
  // V_WMMA_F32_16X16X4_F32 : D = A(16x4,f32) * B(4x16,f32) + C(16x16,f32)
  return __builtin_amdgcn_wmma_f32_16x16x4_f32(false, a, false, b, (short)0, c, false, false);
}
__device__ __forceinline__ v2f ld2(const float* p) { return *(const v2f*)p; }

// ---------- small utility kernels ----------
__global__ void k_clear4(float4* p, int n4) {   // n4 = count of float4
  int i = blockIdx.x * blockDim.x + threadIdx.x;
  if (i < n4) p[i] = make_float4(0.f, 0.f, 0.f, 0.f);
}
__global__ void k_clearu(unsigned* p, int n) {
  int i = blockIdx.x * blockDim.x + threadIdx.x;
  if (i < n) p[i] = 0u;
}
__global__ void k_xn0(float* XN, const float* pm25) {
  int bn = blockIdx.x * blockDim.x + threadIdx.x;
  if (bn >= NB * NN) return;
  int b = bn / NN, n = bn % NN;
  XN[bn] = pm25[((size_t)b * HISTN + (HISTN - 1)) * NN + n];
}

// ---------- weight prep: K-contiguous, zero-padded ----------
__global__ void k_prep_w1t(const float* W1, float* W1t) {   // (2048 rows x 16 k)
  int i = blockIdx.x * blockDim.x + threadIdx.x;
  if (i >= C1 * 16) return;
  int n = i >> 4, k = i & 15;
  W1t[i] = (k < INDIM) ? W1[(size_t)k * C1 + n] : 0.f;
}
__global__ void k_prep_w2t(const float* W2, float* W2t) {   // (16 rows x 2048 k)
  int i = blockIdx.x * blockDim.x + threadIdx.x;
  if (i >= 16 * C1) return;
  int n = i / C1, k = i % C1;
  W2t[i] = (n < GOUT) ? W2[(size_t)k * GOUT + n] : 0.f;
}
__global__ void k_prep_wih(const float* Wih, float* Wp) {   // (192 rows x 28 k)
  int i = blockIdx.x * blockDim.x + threadIdx.x;
  if (i >= 192 * 28) return;
  int n = i / 28, k = i % 28;
  Wp[i] = (k < 27) ? Wih[n * 27 + k] : 0.f;
}

// ---------- build x = concat([xn, feature_t]) into padded (NP x 16) ----------
__global__ void k_build_x(float* X, const float* XN, const float* feat, int t) {
  int bn = blockIdx.x * blockDim.x + threadIdx.x;
  if (bn >= NB * NN) return;
  int b = bn / NN, n = bn % NN;
  float* xp = X + ((size_t)b * NP + n) * 16;
  xp[0] = XN[bn];
  const float* fp = feat + (((size_t)b * (HISTN + PREDN) + HISTN + t) * NN + n) * (INDIM - 1);
  #pragma unroll
  for (int c = 0; c < INDIM - 1; ++c) xp[1 + c] = fp[c];
  xp[14] = 0.f; xp[15] = 0.f;
}

// ---------- GAT1 GEMM: H[b] = X[b](NPx16) @ W1(14x2048); 2 N-tiles/wave ----------
__global__ __launch_bounds__(32) void k_gemm_gat1(const float* __restrict__ X,
                                                  const float* __restrict__ W1t,
                                                  float* __restrict__ H) {
  int mt = blockIdx.x, nt0 = blockIdx.y * 2, b = blockIdx.z;
  int lane = threadIdx.x, half = lane >> 4, l15 = lane & 15;
  const float* Xb = X + (size_t)b * NP * 16;
  float* Hb = H + (size_t)b * NP * C1;
  int arow = mt * 16 + l15;
  int bcol = nt0 * 16 + l15;
  const float* Ap  = Xb + (size_t)arow * 16;
  const float* Bp0 = W1t + (size_t)bcol * 16;
  const float* Bp1 = Bp0 + 16 * 16;
  v8f c0 = {}, c1 = {};
  #pragma unroll
  for (int k0 = 0; k0 < 16; k0 += 4) {
    int ka = k0 + half * 2;
    v2f av = ld2(Ap + ka);
    v2f b0 = ld2(Bp0 + ka);
    v2f b1 = ld2(Bp1 + ka);
    c0 = wmma4(av, b0, c0);
    c1 = wmma4(av, b1, c1);
  }
  int mbase = mt * 16 + half * 8;
  #pragma unroll
  for (int r = 0; r < 8; ++r) {
    Hb[(size_t)(mbase + r) * C1 + bcol] = c0[r];
    Hb[(size_t)(mbase + r) * C1 + bcol + 16] = c1[r];
  }
}

// ---------- per-node attention scores ----------
__global__ void k_scores1(const float* H, const float* a1s, const float* a1d,
                          float* Ss, float* Sd) {
  int idx = blockIdx.x * blockDim.x + threadIdx.x;
  if (idx >= NB * NN * HEADS) return;
  int h = idx % HEADS;
  int bn = idx / HEADS;
  int b = bn / NN, n = bn % NN;
  const float* hp = H + ((size_t)b * NP + n) * C1 + h * HIDG;
  float ss = 0.f, sd = 0.f;
  for (int d = 0; d < HIDG; ++d) {
    float v = hp[d];
    ss += v * a1s[h * HIDG + d];
    sd += v * a1d[h * HIDG + d];
  }
  Ss[idx] = ss; Sd[idx] = sd;
}

// monotonic float<->uint order encoding for atomicMax
__device__ __forceinline__ unsigned fenc(float f) {
  unsigned u = __float_as_uint(f);
  return (u & 0x80000000u) ? ~u : (u | 0x80000000u);
}
__device__ __forceinline__ float fdec(unsigned k) {
  return (k & 0x80000000u) ? __uint_as_float(k ^ 0x80000000u) : __uint_as_float(~k);
}
__device__ __forceinline__ void edge_sd(const int* ei, int E, int e, int& s, int& d) {
  s = (e < E) ? ei[e] : (e - E);
  d = (e < E) ? ei[E + e] : (e - E);
}

__global__ void k_edge1_max(const int* ei, int E, int Etot, const float* Ss,
                            const float* Sd, unsigned* M1) {
  int idx = blockIdx.x * blockDim.x + threadIdx.x;
  if (idx >= NB * Etot) return;
  int b = idx / Etot, e = idx % Etot, s, d;
  edge_sd(ei, E, e, s, d);
  const float* Ssb = Ss + (size_t)b * NN * HEADS;
  const float* Sdb = Sd + (size_t)b * NN * HEADS;
  unsigned* Mb = M1 + (size_t)b * NN * HEADS;
  #pragma unroll
  for (int h = 0; h < HEADS; ++h) {
    float v = Ssb[s * HEADS + h] + Sdb[d * HEADS + h];
    v = v > 0.f ? v : 0.2f * v;                 // leaky_relu 0.2
    atomicMax(&Mb[d * HEADS + h], fenc(v));
  }
}

__global__ void k_edge1_exp(const int* ei, int E, int Etot, const float* Ss,
                            const float* Sd, const unsigned* M1, float* SUM1,
                            float* EX1) {
  int idx = blockIdx.x * blockDim.x + threadIdx.x;
  if (idx >= NB * Etot) return;
  int b = idx / Etot, e = idx % Etot, s, d;
  edge_sd(ei, E, e, s, d);
  const float* Ssb = Ss + (size_t)b * NN * HEADS;
  const float* Sdb = Sd + (size_t)b * NN * HEADS;
  const unsigned* Mb = M1 + (size_t)b * NN * HEADS;
  float* Sumb = SUM1 + (size_t)b * NN * HEADS;
  #pragma unroll
  for (int h = 0; h < HEADS; ++h) {
    float v = Ssb[s * HEADS + h] + Sdb[d * HEADS + h];
    v = v > 0.f ? v : 0.2f * v;
    float ex = expf(v - fdec(Mb[d * HEADS + h]));
    EX1[(size_t)idx * HEADS + h] = ex;
    atomicAdd(&Sumb[d * HEADS + h], ex);
  }
}

// aggregate: OUT1[dst,h,:] += alpha * H[src,h,:]  (block = 256 = 8 heads x 32 lanes)
__global__ __launch_bounds__(256) void k_agg1(const int* ei, int E, int Etot,
                                              const float* EX1, const float* SUM1,
                                              const float* H, float* OUT1) {
  int be = blockIdx.x;
  int b = be / Etot, e = be % Etot, s, d;
  edge_sd(ei, E, e, s, d);
  int h = threadIdx.x >> 5, l = threadIdx.x & 31;
  float alpha = EX1[(size_t)be * HEADS + h] /
                SUM1[((size_t)b * NN + d) * HEADS + h];
  const float* Hs = H + ((size_t)b * NP + s) * C1 + h * HIDG;
  float* Od = OUT1 + ((size_t)b * NP + d) * C1 + h * HIDG;
  #pragma unroll
  for (int j = 0; j < HIDG; j += 32)
    atomicAdd(&Od[j + l], alpha * Hs[j + l]);
}

__global__ void k_elu_bias(float* OUT1, const float* b1) {
  size_t i = (size_t)blockIdx.x * blockDim.x + threadIdx.x;
  if (i >= (size_t)NB * NP * C1) return;
  float v = OUT1[i] + b1[i % C1];
  OUT1[i] = v > 0.f ? v : expm1f(v);
}

// ---------- GAT2 GEMM (K-split x8): H2[b] += OUT1[b](NPx2048) @ W2t ----------
#define KCH (C1 / 8)   /* 256 per K-chunk */
__global__ __launch_bounds__(32) void k_gemm_gat2(const float* __restrict__ A,
                                                  const float* __restrict__ W2t,
                                                  float* __restrict__ H2) {
  int mt = blockIdx.x, kc = blockIdx.y, b = blockIdx.z;
  int lane = threadIdx.x, half = lane >> 4, l15 = lane & 15;
  const float* Ap = A + ((size_t)b * NP + mt * 16 + l15) * C1 + kc * KCH;
  const float* Bp = W2t + (size_t)l15 * C1 + kc * KCH;
  v8f c = {};
  #pragma unroll 4
  for (int k0 = 0; k0 < KCH; k0 += 4) {
    int ka = k0 + half * 2;
    v2f av = ld2(Ap + ka);
    v2f bv = ld2(Bp + ka);
    c = wmma4(av, bv, c);
  }
  float* H2b = H2 + (size_t)b * NP * 16;
  int mbase = mt * 16 + half * 8;
  #pragma unroll
  for (int r = 0; r < 8; ++r)
    atomicAdd(&H2b[(size_t)(mbase + r) * 16 + l15], c[r]);
}

__global__ void k_scores2(const float* H2, const float* a2s, const float* a2d,
                          float* Ss, float* Sd) {
  int bn = blockIdx.x * blockDim.x + threadIdx.x;
  if (bn >= NB * NN) return;
  int b = bn / NN, n = bn % NN;
  const float* hp = H2 + ((size_t)b * NP + n) * 16;
  float ss = 0.f, sd = 0.f;
  #pragma unroll
  for (int d = 0; d < GOUT; ++d) { ss += hp[d] * a2s[d]; sd += hp[d] * a2d[d]; }
  Ss[bn] = ss; Sd[bn] = sd;
}

__global__ void k_edge2_max(const int* ei, int E, int Etot, const float* Ss,
                            const float* Sd, unsigned* M2) {
  int idx = blockIdx.x * blockDim.x + threadIdx.x;
  if (idx >= NB * Etot) return;
  int b = idx / Etot, e = idx % Etot, s, d;
  edge_sd(ei, E, e, s, d);
  float v = Ss[b * NN + s] + Sd[b * NN + d];
  v = v > 0.f ? v : 0.2f * v;
  atomicMax(&M2[b * NN + d], fenc(v));
}

__global__ void k_edge2_exp(const int* ei, int E, int Etot, const float* Ss,
                            const float* Sd, const unsigned* M2, float* SUM2,
                            float* EX2) {
  int idx = blockIdx.x * blockDim.x + threadIdx.x;
  if (idx >= NB * Etot) return;
  int b = idx / Etot, e = idx % Etot, s, d;
  edge_sd(ei, E, e, s, d);
  float v = Ss[b * NN + s] + Sd[b * NN + d];
  v = v > 0.f ? v : 0.2f * v;
  float ex = expf(v - fdec(M2[b * NN + d]));
  EX2[idx] = ex;
  atomicAdd(&SUM2[b * NN + d], ex);
}

__global__ void k_agg2(const int* ei, int E, int Etot, const float* EX2,
                       const float* SUM2, const float* H2, float* G) {
  int idx = blockIdx.x * blockDim.x + threadIdx.x;
  if (idx >= NB * Etot) return;
  int b = idx / Etot, e = idx % Etot, s, d;
  edge_sd(ei, E, e, s, d);
  float alpha = EX2[idx] / SUM2[b * NN + d];
  const float* Hs = H2 + ((size_t)b * NP + s) * 16;
  float* Gd = G + ((size_t)b * NN + d) * 16;
  #pragma unroll
  for (int c = 0; c < GOUT; ++c) atomicAdd(&Gd[c], alpha * Hs[c]);
}

// gx = concat([g + b2, x]) padded to 28 cols (M = 8000 exact)
__global__ void k_build_gx(float* GX, const float* G, const float* X, const float* b2) {
  int bn = blockIdx.x * blockDim.x + threadIdx.x;
  if (bn >= NB * NN) return;
  int b = bn / NN, n = bn % NN;
  float* gp = GX + (size_t)bn * 28;
  const float* g = G + (size_t)bn * 16;
  const float* x = X + ((size_t)b * NP + n) * 16;
  #pragma unroll
  for (int c = 0; c < GOUT; ++c) gp[c] = g[c] + b2[c];
  #pragma unroll
  for (int c = 0; c < INDIM; ++c) gp[GOUT + c] = x[c];
  gp[27] = 0.f;
}

// ---------- GRU GEMM: APRE = GX(8000x28) @ Wihp^T + b_ih; 2 N-tiles/wave ----------
__global__ __launch_bounds__(32) void k_gemm_ih(const float* __restrict__ GX,
                                                const float* __restrict__ Wihp,
                                                const float* __restrict__ bih,
                                                float* __restrict__ APRE) {
  int mt = blockIdx.x, nt0 = blockIdx.y * 2;
  int lane = threadIdx.x, half = lane >> 4, l15 = lane & 15;
  const float* Ap  = GX + (size_t)(mt * 16 + l15) * 28;
  int n0 = nt0 * 16 + l15;
  const float* Bp0 = Wihp + (size_t)n0 * 28;
  const float* Bp1 = Bp0 + 16 * 28;
  v8f c0 = {}, c1 = {};
  #pragma unroll
  for (int k0 = 0; k0 < 28; k0 += 4) {
    int ka = k0 + half * 2;
    v2f av = ld2(Ap + ka);
    c0 = wmma4(av, ld2(Bp0 + ka), c0);
    c1 = wmma4(av, ld2(Bp1 + ka), c1);
  }
  int mbase = mt * 16 + half * 8;
  #pragma unroll
  for (int r = 0; r < 8; ++r) {
    APRE[(size_t)(mbase + r) * 192 + n0] = c0[r] + bih[n0];
    APRE[(size_t)(mbase + r) * 192 + n0 + 16] = c1[r] + bih[n0 + 16];
  }
}

// HPRE = HN(8000x64) @ Whh^T + b_hh; Whh is (192x64) K-contiguous already
__global__ __launch_bounds__(32) void k_gemm_hh(const float* __restrict__ HN,
                                                const float* __restrict__ Whh,
                                                const float* __restrict__ bhh,
                                                float* __restrict__ HPRE) {
  int mt = blockIdx.x, nt0 = blockIdx.y * 2;
  int lane = threadIdx.x, half = lane >> 4, l15 = lane & 15;
  const float* Ap  = HN + (size_t)(mt * 16 + l15) * HID;
  int n0 = nt0 * 16 + l15;
  const float* Bp0 = Whh + (size_t)n0 * HID;
  const float* Bp1 = Bp0 + 16 * HID;
  v8f c0 = {}, c1 = {};
  #pragma unroll
  for (int k0 = 0; k0 < HID; k0 += 4) {
    int ka = k0 + half * 2;
    v2f av = ld2(Ap + ka);
    c0 = wmma4(av, ld2(Bp0 + ka), c0);
    c1 = wmma4(av, ld2(Bp1 + ka), c1);
  }
  int mbase = mt * 16 + half * 8;
  #pragma unroll
  for (int r = 0; r < 8; ++r) {
    HPRE[(size_t)(mbase + r) * 192 + n0] = c0[r] + bhh[n0];
    HPRE[(size_t)(mbase + r) * 192 + n0 + 16] = c1[r] + bhh[n0 + 16];
  }
}

// GRU gates + fc_out (one 64-thread block per row, LDS reduction for fc)
__global__ __launch_bounds__(64) void k_gru(const float* APRE, const float* HPRE,
                                            const float* HNold, float* HNnew,
                                            float* XNnew, float* out,
                                            const float* fcW, const float* fcB, int t) {
  int bn = blockIdx.x, j = threadIdx.x;
  const float* a = APRE + (size_t)bn * 192;
  const float* hp = HPRE + (size_t)bn * 192;
  float r = 1.f / (1.f + expf(-(a[j] + hp[j])));
  float z = 1.f / (1.f + expf(-(a[64 + j] + hp[64 + j])));
  float nv = tanhf(a[128 + j] + r * hp[128 + j]);
  float ho = HNold[(size_t)bn * HID + j];
  float hnew = (1.f - z) * nv + z * ho;
  HNnew[(size_t)bn * HID + j] = hnew;
  __shared__ float red[64];
  red[j] = hnew * fcW[j];
  __syncthreads();
  for (int s = 32; s > 0; s >>= 1) { if (j < s) red[j] += red[j + s]; __syncthreads(); }
  if (j == 0) {
    float x = red[0] + fcB[0];
    XNnew[bn] = x;
    int b = bn / NN, n = bn % NN;
    out[((size_t)b * PREDN + t) * NN + n] = x;   // (B, PRED, N, 1)
  }
}

// ==========================================================================
extern "C" void kernel_launch(void* const* d_in, const int* in_sizes, int n_in,
                              void* d_out, int out_size, void* d_ws, size_t ws_size,
                              hipStream_t stream) {
  const float* pm25 = (const float*)d_in[0];
  const float* feat = (const float*)d_in[1];
  const int*   ei   = (const int*)d_in[2];
  const float* W1   = (const float*)d_in[3];
  const float* a1s  = (const float*)d_in[4];
  const float* a1d  = (const float*)d_in[5];
  const float* b1   = (const float*)d_in[6];
  const float* W2   = (const float*)d_in[7];
  const float* a2s  = (const float*)d_in[8];
  const float* a2d  = (const float*)d_in[9];
  const float* b2   = (const float*)d_in[10];
  const float* Wih  = (const float*)d_in[11];
  const float* Whh  = (const float*)d_in[12];
  const float* bih  = (const float*)d_in[13];
  const float* bhh  = (const float*)d_in[14];
  const float* fcW  = (const float*)d_in[15];
  const float* fcB  = (const float*)d_in[16];
  float* out = (float*)d_out;

  const int E = in_sizes[2] / 2;
  const int Etot = E + NN;

  char* ws = (char*)d_ws;
  size_t off = 0;
  auto carve = [&](size_t bytes) -> void* {
    void* p = ws + off;
    off += (bytes + 255) & ~(size_t)255;
    return p;
  };
  float*    X    = (float*)carve((size_t)NB * NP * 16 * 4);
  float*    H    = (float*)carve((size_t)NB * NP * C1 * 4);
  float*    OUT1 = (float*)carve((size_t)NB * NP * C1 * 4);
  float*    S1s  = (float*)carve((size_t)NB * NN * HEADS * 4);
  float*    S1d  = (float*)carve((size_t)NB * NN * HEADS * 4);
  unsigned* M1   = (unsigned*)carve((size_t)NB * NN * HEADS * 4);
  float*    SUM1 = (float*)carve((size_t)NB * NN * HEADS * 4);
  float*    EX1  = (float*)carve((size_t)NB * Etot * HEADS * 4);
  float*    H2   = (float*)carve((size_t)NB * NP * 16 * 4);
  float*    S2s  = (float*)carve((size_t)NB * NN * 4);
  float*    S2d  = (float*)carve((size_t)NB * NN * 4);
  unsigned* M2   = (unsigned*)carve((size_t)NB * NN * 4);
  float*    SUM2 = (float*)carve((size_t)NB * NN * 4);
  float*    EX2  = (float*)carve((size_t)NB * Etot * 4);
  float*    G    = (float*)carve((size_t)NB * NN * 16 * 4);
  float*    GX   = (float*)carve((size_t)NB * NN * 28 * 4);
  float*    APRE = (float*)carve((size_t)NB * NN * 192 * 4);
  float*    HPRE = (float*)carve((size_t)NB * NN * 192 * 4);
  float*    HN0  = (float*)carve((size_t)NB * NN * HID * 4);
  float*    HN1  = (float*)carve((size_t)NB * NN * HID * 4);
  float*    XN0  = (float*)carve((size_t)NB * NN * 4);
  float*    XN1  = (float*)carve((size_t)NB * NN * 4);
  float*    W1t  = (float*)carve((size_t)C1 * 16 * 4);
  float*    W2t  = (float*)carve((size_t)16 * C1 * 4);
  float*    Wihp = (float*)carve((size_t)192 * 28 * 4);
  (void)ws_size; (void)n_in; (void)out_size;

  const int BN = NB * NN;                  // 8000
  const int T256 = 256;
  auto blk  = [](int n) { return (n + 255) / 256; };
  auto blk4 = [](int n) { return (n / 4 + 255) / 256; };   // for k_clear4

  // one-time per launch: weight prep, zero the padded X, initial state
  k_prep_w1t<<<blk(C1 * 16), T256, 0, stream>>>(W1, W1t);
  k_prep_w2t<<<blk(16 * C1), T256, 0, stream>>>(W2, W2t);
  k_prep_wih<<<blk(192 * 28), T256, 0, stream>>>(Wih, Wihp);
  k_clear4<<<blk4(NB * NP * 16), T256, 0, stream>>>((float4*)X, NB * NP * 16 / 4);
  k_clear4<<<blk4(BN * HID), T256, 0, stream>>>((float4*)HN0, BN * HID / 4);
  k_xn0<<<blk(BN), T256, 0, stream>>>(XN0, pm25);

  for (int t = 0; t < PREDN; ++t) {
    float* hn_in  = (t & 1) ? HN1 : HN0;
    float* hn_out = (t & 1) ? HN0 : HN1;
    float* xn_in  = (t & 1) ? XN1 : XN0;
    float* xn_out = (t & 1) ? XN0 : XN1;

    k_build_x<<<blk(BN), T256, 0, stream>>>(X, xn_in, feat, t);

    // GAT layer 1
    k_gemm_gat1<<<dim3(NP / 16, C1 / 32, NB), 32, 0, stream>>>(X, W1t, H);
    k_scores1<<<blk(BN * HEADS), T256, 0, stream>>>(H, a1s, a1d, S1s, S1d);
    k_clearu<<<blk(BN * HEADS), T256, 0, stream>>>(M1, BN * HEADS);
    k_clear4<<<blk4(BN * HEADS), T256, 0, stream>>>((float4*)SUM1, BN * HEADS / 4);
    k_clear4<<<blk4(NB * NP * C1), T256, 0, stream>>>((float4*)OUT1, NB * NP * C1 / 4);
    k_edge1_max<<<blk(NB * Etot), T256, 0, stream>>>(ei, E, Etot, S1s, S1d, M1);
    k_edge1_exp<<<blk(NB * Etot), T256, 0, stream>>>(ei, E, Etot, S1s, S1d, M1, SUM1, EX1);
    k_agg1<<<NB * Etot, 256, 0, stream>>>(ei, E, Etot, EX1, SUM1, H, OUT1);
    k_elu_bias<<<blk(NB * NP * C1), T256, 0, stream>>>(OUT1, b1);

    // GAT layer 2 (K split 8 ways, atomic accumulate into cleared H2)
    k_clear4<<<blk4(NB * NP * 16), T256, 0, stream>>>((float4*)H2, NB * NP * 16 / 4);
    k_gemm_gat2<<<dim3(NP / 16, 8, NB), 32, 0, stream>>>(OUT1, W2t, H2);
    k_scores2<<<blk(BN), T256, 0, stream>>>(H2, a2s, a2d, S2s, S2d);
    k_clearu<<<blk(BN), T256, 0, stream>>>(M2, BN);
    k_clear4<<<blk4(BN), T256, 0, stream>>>((float4*)SUM2, BN / 4);
    k_clear4<<<blk4(BN * 16), T256, 0, stream>>>((float4*)G, BN * 16 / 4);
    k_edge2_max<<<blk(NB * Etot), T256, 0, stream>>>(ei, E, Etot, S2s, S2d, M2);
    k_edge2_exp<<<blk(NB * Etot), T256, 0, stream>>>(ei, E, Etot, S2s, S2d, M2, SUM2, EX2);
    k_agg2<<<blk(NB * Etot), T256, 0, stream>>>(ei, E, Etot, EX2, SUM2, H2, G);

    // GRU step + prediction
    k_build_gx<<<blk(BN), T256, 0, stream>>>(GX, G, X, b2);
    k_gemm_ih<<<dim3(BN / 16, 192 / 32), 32, 0, stream>>>(GX, Wihp, bih, APRE);
    k_gemm_hh<<<dim3(BN / 16, 192 / 32), 32, 0, stream>>>(hn_in, Whh, bhh, HPRE);
    k_gru<<<BN, HID, 0, stream>>>(APRE, HPRE, hn_in, hn_out, xn_out, out, fcW, fcB, t);
  }
}